// VectorQuantizer_9706626089941
// MI455X (gfx1250) — compile-verified
//
#include <hip/hip_runtime.h>
#include <math.h>
#include <stdint.h>

typedef float v2f __attribute__((ext_vector_type(2)));
typedef float v8f __attribute__((ext_vector_type(8)));
typedef unsigned int v4u __attribute__((ext_vector_type(4)));
typedef int v4i __attribute__((ext_vector_type(4)));
typedef int v8i __attribute__((ext_vector_type(8)));

#define CB_K        8192
#define CDIM        64
#define NTOK        32768          // 8*64*64 tokens
#define HW          4096           // 64*64
#define QOUT_ELEMS  2097152        // 8*64*4096
#define NTILE       (CB_K / 16)    // 512 codebook tiles
#define LDS_PAD     68             // 64-float row + 4 dwords pad -> conflict-free strided reads
#define TILE_F      (16 * LDS_PAD) // floats per staged tile
#define TILE_BYTES  (TILE_F * 4)

// HAVE_TDM is evaluated per compilation pass: the AMDGCN builtins are only
// declared on the device pass, so the host pass parses the fallback branch
// (discarded) while the device pass must take the TDM path.
#if __has_builtin(__builtin_amdgcn_tensor_load_to_lds) && __has_builtin(__builtin_amdgcn_s_wait_tensorcnt)
#define HAVE_TDM 1
#else
#define HAVE_TDM 0
#endif

#if defined(__HIP_DEVICE_COMPILE__) && !HAVE_TDM
#error "device pass lacks TDM builtins (__builtin_amdgcn_tensor_load_to_lds / __builtin_amdgcn_s_wait_tensorcnt)"
#endif

// d_ws layout (floats):
//   [0 .. 8192)        cnorm[k] = ||codebook[k]||^2
//   [8192 .. 16384)    counts[k] (uint bits)
//   [16384]            sumsq accumulator

// ---------------------------------------------------------------- prep ----
__global__ void vq_prep(const float* __restrict__ cb, float* __restrict__ ws) {
    int i = blockIdx.x * blockDim.x + threadIdx.x;
    if (i < CB_K) {
        const float4* row = (const float4*)(cb + (size_t)i * CDIM);
        float s = 0.f;
        #pragma unroll
        for (int c = 0; c < CDIM / 4; ++c) {
            float4 v = row[c];
            s += v.x * v.x + v.y * v.y + v.z * v.z + v.w * v.w;
        }
        ws[i] = s;
        ((unsigned int*)ws)[CB_K + i] = 0u;   // zero histogram every launch
        if (i == 0) ws[2 * CB_K] = 0.f;       // zero sumsq every launch
    }
}

#if HAVE_TDM
// Issue a TDM load of one 16-code x 64-dim f32 codebook tile into LDS.
// D# per CDNA5 ISA:
//   group0: count=1 | lds_addr | global_addr[56:0] | type=2
//   group1: data_size=4B, pad_enable, pad_interval=5 (pad after 32 x 8B = one
//           64-float row), pad_amount=3 (4 dwords) -> LDS row stride 68 floats,
//           tensor 64 x 8192 (stride 64), tile 64 x 16.
__device__ __forceinline__ void tdm_load_tile(const float* gsrc, unsigned lds_byte_addr) {
    unsigned long long ga = (unsigned long long)(uintptr_t)gsrc;
    v4u g0;
    g0.x = 1u;                                                   // count=1 (valid)
    g0.y = lds_byte_addr;                                        // lds_addr
    g0.z = (unsigned)(ga & 0xffffffffu);                         // global_addr[31:0]
    g0.w = (unsigned)((ga >> 32) & 0x01ffffffu) | 0x80000000u;   // addr[56:32] | type=2<<30
    v8i g1;
    g1[0] = (int)((2u << 16) | (1u << 20) | (5u << 22) | (3u << 25)); // dsize=4B, pad cfg
    g1[1] = (int)(64u << 16);     // tensor_dim0 = 64 (bits 79:48 low half)
    g1[2] = (int)(8192u << 16);   // tensor_dim0 hi=0 ; tensor_dim1 low16 = 8192
    g1[3] = (int)(64u << 16);     // tensor_dim1 hi=0 ; tile_dim0 = 64
    g1[4] = 16;                   // tile_dim1 = 16 ; tile_dim2 = 0
    g1[5] = 64;                   // tensor_dim0_stride = 64
    g1[6] = 0;
    g1[7] = 0;
    v4i z4 = {0, 0, 0, 0};
#if __clang_major__ >= 23
    v8i z8 = {0, 0, 0, 0, 0, 0, 0, 0};
    __builtin_amdgcn_tensor_load_to_lds(g0, g1, z4, z4, z8, 0);
#else
    __builtin_amdgcn_tensor_load_to_lds(g0, g1, z4, z4, 0);
#endif
}
#endif

// ------------------------------------------------------------- argmin ----
// 4 waves/block, 32 tokens per wave (two 16x16 accumulators sharing each B
// fragment), 128 tokens per block. D = cnorm - 2 * z . c via f32 WMMA.
// Codebook tiles double-buffered in LDS via the Tensor Data Mover; one
// barrier per tile; cnorm software-pipelined one tile ahead.
__global__ __launch_bounds__(128) void vq_argmin(const float* __restrict__ z,
                                                 const float* __restrict__ cb,
                                                 const float* __restrict__ ws,
                                                 float* __restrict__ tok_out) {
    __shared__ float btile[2][TILE_F];

    const int tid  = threadIdx.x;
    const int wave = tid >> 5;
    const int lane = tid & 31;
    const int half = lane >> 4;     // 0: lanes 0-15, 1: lanes 16-31
    const int l16  = lane & 15;
    const int tokBase = blockIdx.x * 128 + wave * 32;

    // A fragments for two 16-token sub-tiles, scaled by -2. ISA layout:
    // lane -> M = lane%16 ; VGPR j -> K = 2*(lane/16) + j  (per 4-wide K step)
    v2f a0[16], a1[16];
    {
        int n0 = tokBase + l16;
        int n1 = tokBase + 16 + l16;
        const float* zp0 = z + (size_t)(n0 >> 12) * ((size_t)CDIM * HW) + (n0 & (HW - 1));
        const float* zp1 = z + (size_t)(n1 >> 12) * ((size_t)CDIM * HW) + (n1 & (HW - 1));
        #pragma unroll
        for (int kk = 0; kk < 16; ++kk) {
            int d0 = kk * 4 + 2 * half;
            a0[kk].x = -2.0f * zp0[(size_t)d0 * HW];
            a0[kk].y = -2.0f * zp0[(size_t)(d0 + 1) * HW];
            a1[kk].x = -2.0f * zp1[(size_t)d0 * HW];
            a1[kk].y = -2.0f * zp1[(size_t)(d0 + 1) * HW];
        }
    }

    float bv0[8], bv1[8];
    int   bi0[8], bi1[8];
    #pragma unroll
    for (int v = 0; v < 8; ++v) {
        bv0[v] = 3.4e38f; bi0[v] = 0;
        bv1[v] = 3.4e38f; bi1[v] = 0;
    }

#if HAVE_TDM
    unsigned ldsbase = (unsigned)(uintptr_t)&btile[0][0];
    if (wave == 0) tdm_load_tile(cb, ldsbase);      // prologue: tile 0 -> buf 0
#endif
    float cn_cur = ws[l16];                         // cnorm for tile 0

    for (int t = 0; t < NTILE; ++t) {
        const int cb0 = t * 16;

#if HAVE_TDM
        if (wave == 0) __builtin_amdgcn_s_wait_tensorcnt(0);   // tile t in LDS
        __syncthreads();                                       // publish; prior buf reads done
        if (wave == 0 && t + 1 < NTILE)                        // DMA tile t+1 over buf read at t-1
            tdm_load_tile(cb + (size_t)(cb0 + 16) * CDIM,
                          ldsbase + (unsigned)((t + 1) & 1) * TILE_BYTES);
#else
        // host-pass-only fallback (device pass always has TDM; see #error above)
        __syncthreads();
        {
            float* btw = &btile[t & 1][0];
            #pragma unroll
            for (int r = 0; r < 2; ++r) {
                int idx = tid + 128 * r;              // 0..255 float4 chunks
                int cc  = idx >> 4;                   // code row 0..15
                int dd  = (idx & 15) * 4;             // dim 0..63 step 4
                *(float4*)&btw[cc * LDS_PAD + dd] =
                    *(const float4*)&cb[(size_t)(cb0 + cc) * CDIM + dd];
            }
        }
        __syncthreads();
#endif

        // prefetch next tile's cnorm (t=511 touches counts region of ws: in
        // bounds, value never consumed)
        float cn_next = ws[cb0 + 16 + l16];

        const float* bt = &btile[t & 1][0];
        v8f acc0 = {cn_cur, cn_cur, cn_cur, cn_cur, cn_cur, cn_cur, cn_cur, cn_cur};
        v8f acc1 = acc0;

        #pragma unroll
        for (int kk = 0; kk < 16; ++kk) {
            v2f bf;                             // B: lane -> N = lane%16, VGPR j -> K
            int d0 = kk * 4 + 2 * half;
            bf.x = bt[l16 * LDS_PAD + d0];
            bf.y = bt[l16 * LDS_PAD + d0 + 1];
            acc0 = __builtin_amdgcn_wmma_f32_16x16x4_f32(
                false, a0[kk], false, bf, (short)0, acc0, false, false);
            acc1 = __builtin_amdgcn_wmma_f32_16x16x4_f32(
                false, a1[kk], false, bf, (short)0, acc1, false, false);
        }

        int code = cb0 + l16;
        #pragma unroll
        for (int v = 0; v < 8; ++v) {
            float d0v = acc0[v];
            float d1v = acc1[v];
            if (d0v < bv0[v]) { bv0[v] = d0v; bi0[v] = code; }
            if (d1v < bv1[v]) { bv1[v] = d1v; bi1[v] = code; }
        }

        cn_cur = cn_next;
    }

    // reduce argmin across the 16 lanes of each half (xor masks stay in-half);
    // ties -> smaller code index (matches jnp.argmin first-occurrence)
    #pragma unroll
    for (int v = 0; v < 8; ++v) {
        float b0 = bv0[v]; int i0 = bi0[v];
        float b1 = bv1[v]; int i1 = bi1[v];
        #pragma unroll
        for (int off = 1; off < 16; off <<= 1) {
            float o0 = __shfl_xor(b0, off, 32);
            int   j0 = __shfl_xor(i0, off, 32);
            float o1 = __shfl_xor(b1, off, 32);
            int   j1 = __shfl_xor(i1, off, 32);
            if (o0 < b0 || (o0 == b0 && j0 < i0)) { b0 = o0; i0 = j0; }
            if (o1 < b1 || (o1 == b1 && j1 < i1)) { b1 = o1; i1 = j1; }
        }
        if (l16 == 0) {
            int m = v + 8 * half;               // D row: M = v + 8*(lane/16)
            tok_out[tokBase + m]      = (float)i0;
            tok_out[tokBase + 16 + m] = (float)i1;
        }
    }
}

// -------------------------------------------------- gather + loss + hist ----
__global__ void vq_gather(const float* __restrict__ z, const float* __restrict__ cb,
                          const float* __restrict__ tok_f, float* __restrict__ qout,
                          float* __restrict__ ws) {
    __shared__ float red[256];
    int n = blockIdx.x * blockDim.x + threadIdx.x;   // token id
    int t = (int)tok_f[n];
    const float4* crow = (const float4*)(cb + (size_t)t * CDIM);
    size_t zbase = (size_t)(n >> 12) * ((size_t)CDIM * HW) + (n & (HW - 1));

    float s = 0.f;
    #pragma unroll
    for (int c4 = 0; c4 < CDIM / 4; ++c4) {
        float4 q = crow[c4];
        size_t base = zbase + (size_t)(c4 * 4) * HW;
        float d0 = q.x - z[base];
        float d1 = q.y - z[base + HW];
        float d2 = q.z - z[base + 2 * HW];
        float d3 = q.w - z[base + 3 * HW];
        s += d0 * d0 + d1 * d1 + d2 * d2 + d3 * d3;
        qout[base]          = q.x;               // quantized_st == quantized
        qout[base + HW]     = q.y;
        qout[base + 2 * HW] = q.z;
        qout[base + 3 * HW] = q.w;
    }
    atomicAdd((unsigned int*)(ws + CB_K) + t, 1u);

    red[threadIdx.x] = s;
    __syncthreads();
    for (int st = 128; st > 0; st >>= 1) {
        if (threadIdx.x < st) red[threadIdx.x] += red[threadIdx.x + st];
        __syncthreads();
    }
    if (threadIdx.x == 0) atomicAdd(ws + 2 * CB_K, red[0]);
}

// ------------------------------------------------------------ finalize ----
__global__ void vq_final(const float* __restrict__ ws, float* __restrict__ out) {
    __shared__ float red[256];
    const unsigned int* counts = (const unsigned int*)(ws + CB_K);
    float s = 0.f;
    for (int i = threadIdx.x; i < CB_K; i += 256) {
        float p = (float)counts[i] / (float)NTOK;
        s += p * logf(p + 1e-10f);
    }
    red[threadIdx.x] = s;
    __syncthreads();
    for (int st = 128; st > 0; st >>= 1) {
        if (threadIdx.x < st) red[threadIdx.x] += red[threadIdx.x + st];
        __syncthreads();
    }
    if (threadIdx.x == 0) {
        float mse = ws[2 * CB_K] / (float)QOUT_ELEMS;
        out[QOUT_ELEMS + NTOK]     = mse * 1.25f;     // loss = (1 + beta) * mse
        out[QOUT_ELEMS + NTOK + 1] = expf(-red[0]);   // perplexity
    }
}

extern "C" void kernel_launch(void* const* d_in, const int* in_sizes, int n_in,
                              void* d_out, int out_size, void* d_ws, size_t ws_size,
                              hipStream_t stream) {
    const float* z  = (const float*)d_in[0];     // [8,64,64,64]
    const float* cb = (const float*)d_in[1];     // [8192,64]
    float* out  = (float*)d_out;
    float* ws   = (float*)d_ws;
    float* tokf = out + QOUT_ELEMS;              // tokens stored as float values

    vq_prep  <<<CB_K / 256, 256, 0, stream>>>(cb, ws);
    vq_argmin<<<NTOK / 128, 128, 0, stream>>>(z, cb, ws, tokf);
    vq_gather<<<NTOK / 256, 256, 0, stream>>>(z, cb, tokf, out, ws);
    vq_final <<<1, 256, 0, stream>>>(ws, out);
}